// Supernetwork_82317343195358
// MI455X (gfx1250) — compile-verified
//
#include <hip/hip_runtime.h>
#include <math.h>

typedef float v2f __attribute__((ext_vector_type(2)));
typedef float v8f __attribute__((ext_vector_type(8)));

#define NB    2048         // batch
#define NBEAM 2
#define NROWS (NB*NBEAM)   // 4096
#define HID   128
#define NCLS  100

// ---------------------------------------------------------------------------
// score pass-through copy
// ---------------------------------------------------------------------------
__global__ void copy_scores_kernel(const float* __restrict__ sv, float* __restrict__ dst) {
    int i = blockIdx.x * 256 + threadIdx.x;
    if (i < NROWS) dst[i] = sv[i];
}

// ---------------------------------------------------------------------------
// conv (VALID 3x3) -> per-(channel,block) partial sum / sumsq  (deterministic)
// one block per image; input image staged in LDS
// ---------------------------------------------------------------------------
template<int IC, int IH, int IW, int OH, int OW>
__global__ void conv_stats_kernel(const float* __restrict__ in,
                                  const float* __restrict__ w,   // [32,IC,3,3]
                                  const float* __restrict__ cb,  // [32]
                                  float* __restrict__ psum,      // [32,2048]
                                  float* __restrict__ psq) {     // [32,2048]
    __shared__ float img[IC * IH * IW];
    __shared__ float red[256];
    __shared__ float red2[256];
    int n = blockIdx.x, tid = threadIdx.x;
    const float* xin = in + (size_t)n * IC * IH * IW;
    for (int i = tid; i < IC * IH * IW; i += 256) img[i] = xin[i];
    __syncthreads();
    for (int c = 0; c < 32; ++c) {
        float s = 0.f, s2 = 0.f;
        const float* wc = w + c * IC * 9;
        float bias = cb[c];
        for (int p = tid; p < OH * OW; p += 256) {
            int oy = p / OW, ox = p % OW;
            float acc = bias;
            for (int ic = 0; ic < IC; ++ic) {
                const float* ip = img + ic * IH * IW + oy * IW + ox;
                const float* wp = wc + ic * 9;
                #pragma unroll
                for (int ky = 0; ky < 3; ++ky)
                    #pragma unroll
                    for (int kx = 0; kx < 3; ++kx)
                        acc = fmaf(ip[ky * IW + kx], wp[ky * 3 + kx], acc);
            }
            s += acc; s2 += acc * acc;
        }
        red[tid] = s; red2[tid] = s2; __syncthreads();
        for (int off = 128; off > 0; off >>= 1) {
            if (tid < off) { red[tid] += red[tid + off]; red2[tid] += red2[tid + off]; }
            __syncthreads();
        }
        if (tid == 0) { psum[c * NB + n] = red[0]; psq[c * NB + n] = red2[0]; }
        __syncthreads();
    }
}

// ---------------------------------------------------------------------------
// reduce per-block partials over 2048 images -> BN scale/shift per channel
// ---------------------------------------------------------------------------
__global__ void bn_reduce_finalize_kernel(const float* __restrict__ psum,
                                          const float* __restrict__ psq,
                                          const float* __restrict__ g,
                                          const float* __restrict__ bb,
                                          float invcount,
                                          float* __restrict__ scale,
                                          float* __restrict__ shift) {
    __shared__ float red[256];
    __shared__ float red2[256];
    int c = blockIdx.x, tid = threadIdx.x;
    float s = 0.f, s2 = 0.f;
    for (int i = tid; i < NB; i += 256) { s += psum[c * NB + i]; s2 += psq[c * NB + i]; }
    red[tid] = s; red2[tid] = s2; __syncthreads();
    for (int off = 128; off > 0; off >>= 1) {
        if (tid < off) { red[tid] += red[tid + off]; red2[tid] += red2[tid + off]; }
        __syncthreads();
    }
    if (tid == 0) {
        float m = red[0] * invcount;
        float v = red2[0] * invcount - m * m;
        float sc = g[c] * rsqrtf(v + 1e-5f);
        scale[c] = sc;
        shift[c] = bb[c] - m * sc;
    }
}

// ---------------------------------------------------------------------------
// recompute conv, apply BN(scale,shift), ReLU, 2x2 avg-pool
// REPEAT=true: write final 128-feature vector twice (beam repeat) into [4096,128]
// ---------------------------------------------------------------------------
template<int IC, int IH, int IW, int PH, int PW, bool REPEAT>
__global__ void conv_bn_pool_kernel(const float* __restrict__ in,
                                    const float* __restrict__ w,
                                    const float* __restrict__ cb,
                                    const float* __restrict__ scale,
                                    const float* __restrict__ shift,
                                    float* __restrict__ out) {
    __shared__ float img[IC * IH * IW];
    int n = blockIdx.x, tid = threadIdx.x;
    const float* xin = in + (size_t)n * IC * IH * IW;
    for (int i = tid; i < IC * IH * IW; i += 256) img[i] = xin[i];
    __syncthreads();
    const int TOT = 32 * PH * PW;
    for (int t = tid; t < TOT; t += 256) {
        int c = t / (PH * PW);
        int rem = t % (PH * PW);
        int py = rem / PW, px = rem % PW;
        const float* wc = w + c * IC * 9;
        float sc = scale[c], sh = shift[c], bias = cb[c];
        float acc4 = 0.f;
        #pragma unroll
        for (int dy = 0; dy < 2; ++dy)
            #pragma unroll
            for (int dx = 0; dx < 2; ++dx) {
                int oy = py * 2 + dy, ox = px * 2 + dx;
                float acc = bias;
                for (int ic = 0; ic < IC; ++ic) {
                    const float* ip = img + ic * IH * IW + oy * IW + ox;
                    const float* wp = wc + ic * 9;
                    #pragma unroll
                    for (int ky = 0; ky < 3; ++ky)
                        #pragma unroll
                        for (int kx = 0; kx < 3; ++kx)
                            acc = fmaf(ip[ky * IW + kx], wp[ky * 3 + kx], acc);
                }
                acc4 += fmaxf(fmaf(acc, sc, sh), 0.f);
            }
        acc4 *= 0.25f;
        if (REPEAT) {
            // TOT==128 here; t == c*4 + py*2 + px == flattened feature index
            out[(size_t)(2 * n) * HID + t]     = acc4;
            out[(size_t)(2 * n + 1) * HID + t] = acc4;
        } else {
            out[((size_t)n * 32 + c) * (PH * PW) + rem] = acc4;
        }
    }
}

// ---------------------------------------------------------------------------
// routed MLP layer via V_WMMA_F32_16X16X4_F32
// grid: 256 m-tiles (16 rows each); block: 256 threads = 8 waves; wave w -> n-tile w
// computes all 4 experts for the tile, selects per-row by argmax(trajectories)
// ---------------------------------------------------------------------------
__global__ void mlp_wmma_kernel(const float* __restrict__ hsrc,  // [4096,128]
                                const float* __restrict__ eW,    // [4,128,128] (e,o,d)
                                const float* __restrict__ eb,    // [4,128]
                                const float* __restrict__ traj,  // [4096,4]
                                float* __restrict__ hdst,        // [4096,128]
                                int relu) {
    __shared__ int sel_s[16];
    int lane   = threadIdx.x & 31;
    int wave   = threadIdx.x >> 5;   // n-tile 0..7
    int m_tile = blockIdx.x;         // 0..255
    if (threadIdx.x < 16) {
        const float* t = traj + (size_t)(m_tile * 16 + threadIdx.x) * 4;
        float best = t[0]; int bi = 0;
        #pragma unroll
        for (int e = 1; e < 4; ++e) { float v = t[e]; if (v > best) { best = v; bi = e; } }
        sel_s[threadIdx.x] = bi;
    }
    __syncthreads();

    int rr    = lane & 15;
    int khalf = (lane >> 4) << 1;          // 0 or 2
    int gn    = wave * 16 + rr;            // output feature (column)
    const float* arow = hsrc + (size_t)(m_tile * 16 + rr) * HID;
    const float* wcolb = eW + (size_t)gn * HID;

    v8f acc0 = {}, acc1 = {}, acc2 = {}, acc3 = {};
    for (int k0 = 0; k0 < HID; k0 += 4) {
        int kb = k0 + khalf;
        v2f a;  a.x = arow[kb];  a.y = arow[kb + 1];
        v2f b0, b1, b2, b3;
        const float* wc = wcolb + kb;
        b0.x = wc[0];                 b0.y = wc[1];
        b1.x = wc[1 * HID * HID];     b1.y = wc[1 * HID * HID + 1];
        b2.x = wc[2 * HID * HID];     b2.y = wc[2 * HID * HID + 1];
        b3.x = wc[3 * HID * HID];     b3.y = wc[3 * HID * HID + 1];
        acc0 = __builtin_amdgcn_wmma_f32_16x16x4_f32(false, a, false, b0, (short)0, acc0, false, false);
        acc1 = __builtin_amdgcn_wmma_f32_16x16x4_f32(false, a, false, b1, (short)0, acc1, false, false);
        acc2 = __builtin_amdgcn_wmma_f32_16x16x4_f32(false, a, false, b2, (short)0, acc2, false, false);
        acc3 = __builtin_amdgcn_wmma_f32_16x16x4_f32(false, a, false, b3, (short)0, acc3, false, false);
    }

    int rowoff = (lane >> 4) << 3;   // 0 or 8 (C/D layout: lanes 16-31 hold rows r+8)
    #pragma unroll
    for (int r = 0; r < 8; ++r) {
        int row = r + rowoff;
        int s = sel_s[row];
        float v = (s == 0) ? acc0[r] : (s == 1) ? acc1[r] : (s == 2) ? acc2[r] : acc3[r];
        v += eb[s * HID + gn];
        if (relu) v = fmaxf(v, 0.f);
        hdst[(size_t)(m_tile * 16 + row) * HID + gn] = v;
    }
}

// ---------------------------------------------------------------------------
// output layer preds = h @ out_W^T + out_b via WMMA; N padded 100 -> 112 (7 tiles)
// block: 224 threads = 7 waves
// ---------------------------------------------------------------------------
__global__ void out_wmma_kernel(const float* __restrict__ hsrc,  // [4096,128]
                                const float* __restrict__ oW,    // [100,128]
                                const float* __restrict__ ob,    // [100]
                                float* __restrict__ preds) {     // [4096,100]
    int lane   = threadIdx.x & 31;
    int wave   = threadIdx.x >> 5;   // 0..6
    int m_tile = blockIdx.x;
    int rr     = lane & 15;
    int khalf  = (lane >> 4) << 1;
    int gn     = wave * 16 + rr;     // 0..111
    bool valid = gn < NCLS;
    const float* arow = hsrc + (size_t)(m_tile * 16 + rr) * HID;
    const float* wrow = oW + (size_t)(valid ? gn : 0) * HID;

    v8f acc = {};
    for (int k0 = 0; k0 < HID; k0 += 4) {
        int kb = k0 + khalf;
        v2f a; a.x = arow[kb]; a.y = arow[kb + 1];
        v2f b; b.x = valid ? wrow[kb] : 0.f; b.y = valid ? wrow[kb + 1] : 0.f;
        acc = __builtin_amdgcn_wmma_f32_16x16x4_f32(false, a, false, b, (short)0, acc, false, false);
    }
    if (valid) {
        float bias = ob[gn];
        int rowoff = (lane >> 4) << 3;
        #pragma unroll
        for (int r = 0; r < 8; ++r) {
            int row = m_tile * 16 + r + rowoff;
            preds[(size_t)row * NCLS + gn] = acc[r] + bias;
        }
    }
}

// ---------------------------------------------------------------------------
// row-wise log_softmax over 100 classes; one 128-thread block per row
// ---------------------------------------------------------------------------
__global__ void logsoftmax_kernel(const float* __restrict__ preds, float* __restrict__ out) {
    __shared__ float red[128];
    int row = blockIdx.x, tid = threadIdx.x;
    const float* p = preds + (size_t)row * NCLS;
    float v = (tid < NCLS) ? p[tid] : -INFINITY;
    red[tid] = v; __syncthreads();
    for (int off = 64; off > 0; off >>= 1) {
        if (tid < off) red[tid] = fmaxf(red[tid], red[tid + off]);
        __syncthreads();
    }
    float mx = red[0]; __syncthreads();
    float e = (tid < NCLS) ? expf(v - mx) : 0.f;
    red[tid] = e; __syncthreads();
    for (int off = 64; off > 0; off >>= 1) {
        if (tid < off) red[tid] += red[tid + off];
        __syncthreads();
    }
    float lse = logf(red[0]) + mx;
    if (tid < NCLS) out[(size_t)row * NCLS + tid] = v - lse;
}

// ---------------------------------------------------------------------------
extern "C" void kernel_launch(void* const* d_in, const int* in_sizes, int n_in,
                              void* d_out, int out_size, void* d_ws, size_t ws_size,
                              hipStream_t stream) {
    const float* x    = (const float*)d_in[0];
    const float* traj = (const float*)d_in[1];
    const float* sv   = (const float*)d_in[2];
    const float* cw1  = (const float*)d_in[3];
    const float* cb1  = (const float*)d_in[4];
    const float* g1   = (const float*)d_in[5];
    const float* bb1  = (const float*)d_in[6];
    const float* cw2  = (const float*)d_in[7];
    const float* cb2  = (const float*)d_in[8];
    const float* g2   = (const float*)d_in[9];
    const float* bb2  = (const float*)d_in[10];
    const float* cw3  = (const float*)d_in[11];
    const float* cb3  = (const float*)d_in[12];
    const float* g3   = (const float*)d_in[13];
    const float* bb3  = (const float*)d_in[14];
    const float* eW   = (const float*)d_in[15];
    const float* eb   = (const float*)d_in[16];
    const float* oW   = (const float*)d_in[17];
    const float* ob   = (const float*)d_in[18];
    float* out = (float*)d_out;
    float* ws  = (float*)d_ws;

    // workspace layout (floats)
    float* sc1 = ws + 0;   float* sh1 = ws + 32;
    float* sc2 = ws + 64;  float* sh2 = ws + 96;
    float* sc3 = ws + 128; float* sh3 = ws + 160;
    float* psum = ws + 256;              // [32,2048]
    float* psq  = psum + 32 * NB;        // [32,2048]
    float* h1   = psq + 32 * NB;         // [2048,32,15,15]
    float* h2   = h1 + (size_t)NB * 32 * 15 * 15;   // [2048,32,6,6]
    float* hA   = h2 + (size_t)NB * 32 * 6 * 6;     // [4096,128]
    float* hB   = hA + (size_t)NROWS * HID;         // [4096,128]
    float* preds= hB + (size_t)NROWS * HID;         // [4096,100]

    // score pass-through (second tuple output)
    copy_scores_kernel<<<16, 256, 0, stream>>>(sv, out + (size_t)NROWS * NCLS);

    // block 1: conv 3->32 on 32x32 -> 30x30, BN stats, BN+ReLU+pool -> 15x15
    conv_stats_kernel<3, 32, 32, 30, 30><<<NB, 256, 0, stream>>>(x, cw1, cb1, psum, psq);
    bn_reduce_finalize_kernel<<<32, 256, 0, stream>>>(psum, psq, g1, bb1,
        1.f / (float)(NB * 30 * 30), sc1, sh1);
    conv_bn_pool_kernel<3, 32, 32, 15, 15, false><<<NB, 256, 0, stream>>>(x, cw1, cb1, sc1, sh1, h1);

    // block 2: conv 32->32 on 15x15 -> 13x13, pool -> 6x6
    conv_stats_kernel<32, 15, 15, 13, 13><<<NB, 256, 0, stream>>>(h1, cw2, cb2, psum, psq);
    bn_reduce_finalize_kernel<<<32, 256, 0, stream>>>(psum, psq, g2, bb2,
        1.f / (float)(NB * 13 * 13), sc2, sh2);
    conv_bn_pool_kernel<32, 15, 15, 6, 6, false><<<NB, 256, 0, stream>>>(h1, cw2, cb2, sc2, sh2, h2);

    // block 3: conv 32->32 on 6x6 -> 4x4, pool -> 2x2, flatten + beam-repeat -> hA[4096,128]
    conv_stats_kernel<32, 6, 6, 4, 4><<<NB, 256, 0, stream>>>(h2, cw3, cb3, psum, psq);
    bn_reduce_finalize_kernel<<<32, 256, 0, stream>>>(psum, psq, g3, bb3,
        1.f / (float)(NB * 4 * 4), sc3, sh3);
    conv_bn_pool_kernel<32, 6, 6, 2, 2, true><<<NB, 256, 0, stream>>>(h2, cw3, cb3, sc3, sh3, hA);

    // routed MLP: 3 depths, WMMA f32, ReLU on first two
    mlp_wmma_kernel<<<256, 256, 0, stream>>>(hA, eW, eb, traj + 0 * NROWS * 4, hB, 1);
    mlp_wmma_kernel<<<256, 256, 0, stream>>>(hB, eW, eb, traj + 1 * NROWS * 4, hA, 1);
    mlp_wmma_kernel<<<256, 256, 0, stream>>>(hA, eW, eb, traj + 2 * NROWS * 4, hB, 0);

    // output layer (WMMA) + log_softmax
    out_wmma_kernel<<<256, 224, 0, stream>>>(hB, oW, ob, preds);
    logsoftmax_kernel<<<NROWS, 128, 0, stream>>>(preds, out);
}